// TemporalGNNBatch_42167988913020
// MI455X (gfx1250) — compile-verified
//
#include <hip/hip_runtime.h>
#include <math.h>

// Problem constants from the reference
#define BB   32
#define NN   51
#define FF   4
#define PP   137
#define EE   408
#define EMB  32
#define ETOT (EE + NN)   // edges + self loops = 459
#define AGG_PITCH 33     // LDS row pitch for WMMA A-operand (conflict-free)

typedef __attribute__((ext_vector_type(2))) float v2f;
typedef __attribute__((ext_vector_type(8))) float v8f;

// ---------------------------------------------------------------------------
// Kernel 1: GCN norm (deg/self-loops), softmax(attention), zero accumulator
// ---------------------------------------------------------------------------
__global__ void prep_kernel(const int* __restrict__ ei,
                            const float* __restrict__ attention,
                            float* __restrict__ hacc,
                            float* __restrict__ normA,
                            int* __restrict__ srcA,
                            int* __restrict__ dstA,
                            float* __restrict__ probs) {
  __shared__ float deg[NN];
  __shared__ float dinv[NN];
  const int t = threadIdx.x;

  if (t < NN) deg[t] = 1.0f;                       // self-loop contribution
  __syncthreads();
  for (int e = t; e < EE; e += blockDim.x)
    atomicAdd(&deg[ei[EE + e]], 1.0f);             // dst-degree
  __syncthreads();
  if (t < NN) dinv[t] = rsqrtf(deg[t]);            // deg >= 1 always
  __syncthreads();

  for (int i = t; i < ETOT; i += blockDim.x) {
    int s, d;
    if (i < EE) { s = ei[i]; d = ei[EE + i]; }
    else        { s = d = i - EE; }                // self loops
    srcA[i] = s; dstA[i] = d;
    normA[i] = dinv[s] * dinv[d];
  }

  // zero the attention-weighted accumulator (re-done every launch)
  for (int i = t; i < BB * NN * EMB; i += blockDim.x) hacc[i] = 0.0f;

  // softmax over 137 attention logits (tiny, serial on one lane)
  if (t == 0) {
    float m = attention[0];
    for (int p = 1; p < PP; ++p) m = fmaxf(m, attention[p]);
    float ssum = 0.0f;
    for (int p = 0; p < PP; ++p) ssum += __expf(attention[p] - m);
    float inv = 1.0f / ssum;
    for (int p = 0; p < PP; ++p) probs[p] = __expf(attention[p] - m) * inv;
  }
}

// ---------------------------------------------------------------------------
// Kernel 2: one block per (p, b). xw -> edge aggregation (LDS atomics) ->
// [64x32]@[32x32] f32 WMMA GEMM per gate -> gates -> probs-weighted atomicAdd
// ---------------------------------------------------------------------------
__global__ __launch_bounds__(128)
void gate_kernel(const float* __restrict__ x,
                 const float* __restrict__ conv_z_w, const float* __restrict__ conv_z_b,
                 const float* __restrict__ conv_h_w, const float* __restrict__ conv_h_b,
                 const float* __restrict__ lin_z_w,  const float* __restrict__ lin_z_b,
                 const float* __restrict__ lin_h_w,  const float* __restrict__ lin_h_b,
                 const float* __restrict__ normA, const int* __restrict__ srcA,
                 const int* __restrict__ dstA,  const float* __restrict__ probs,
                 float* __restrict__ hacc) {
  const int p   = blockIdx.x;
  const int b   = blockIdx.y;
  const int tid = threadIdx.x;

  __shared__ float s_x[NN * FF];
  __shared__ float s_wz[FF * EMB], s_wh[FF * EMB];
  __shared__ float s_czb[EMB], s_chb[EMB];
  __shared__ float s_xwz[NN * EMB], s_xwh[NN * EMB];
  __shared__ float s_aggz[64 * AGG_PITCH], s_aggh[64 * AGG_PITCH];

  // stage x slice (strided gather: x[b,n,f,p], innermost stride = P)
  for (int i = tid; i < NN * FF; i += 128) {
    int n = i >> 2, f = i & 3;
    s_x[i] = x[((b * NN + n) * FF + f) * PP + p];
  }
  for (int i = tid; i < FF * EMB; i += 128) { s_wz[i] = conv_z_w[i]; s_wh[i] = conv_h_w[i]; }
  if (tid < EMB) { s_czb[tid] = conv_z_b[tid]; s_chb[tid] = conv_h_b[tid]; }
  for (int i = tid; i < 64 * AGG_PITCH; i += 128) { s_aggz[i] = 0.0f; s_aggh[i] = 0.0f; }
  __syncthreads();

  // xw = x @ W  (F=4 inner dim; pure VALU)
  for (int i = tid; i < NN * EMB; i += 128) {
    int n = i >> 5, e = i & 31;
    float az = 0.0f, ah = 0.0f;
#pragma unroll
    for (int f = 0; f < FF; ++f) {
      float v = s_x[n * FF + f];
      az += v * s_wz[f * EMB + e];
      ah += v * s_wh[f * EMB + e];
    }
    s_xwz[i] = az; s_xwh[i] = ah;
  }
  __syncthreads();

  // normalized scatter-add over 459 edges (ds_add_f32)
  for (int i = tid; i < ETOT * EMB; i += 128) {
    int ed = i >> 5, e = i & 31;
    int s = srcA[ed], d = dstA[ed];
    float w = normA[ed];
    atomicAdd(&s_aggz[d * AGG_PITCH + e], w * s_xwz[s * EMB + e]);
    atomicAdd(&s_aggh[d * AGG_PITCH + e], w * s_xwh[s * EMB + e]);
  }
  __syncthreads();

  // conv bias (only valid 51 rows; padded rows stay 0 and are masked on store)
  for (int i = tid; i < NN * EMB; i += 128) {
    int n = i >> 5, e = i & 31;
    s_aggz[n * AGG_PITCH + e] += s_czb[e];
    s_aggh[n * AGG_PITCH + e] += s_chb[e];
  }
  __syncthreads();

  // ---- f32 WMMA GEMM: [64x32] @ [32x32] per gate (H=0 -> only top half of lin_*_w)
  const int lane = tid & 31;
  const int wave = tid >> 5;           // 4 waves -> 4 M-tiles of 16 rows
  const int mt   = wave;
  const int lhi  = lane >> 4;          // 0: lanes 0-15, 1: lanes 16-31
  const int ll   = lane & 15;
  const float pw = probs[p];

#pragma unroll
  for (int nt = 0; nt < 2; ++nt) {
    v8f accZ = {0.f, 0.f, 0.f, 0.f, 0.f, 0.f, 0.f, 0.f};
    v8f accH = {0.f, 0.f, 0.f, 0.f, 0.f, 0.f, 0.f, 0.f};
    const int col = nt * 16 + ll;      // B/C/D column for this lane
#pragma unroll
    for (int k0 = 0; k0 < EMB; k0 += 4) {
      // A 16x4 f32 layout: lanes 0-15 -> K = k0,k0+1 ; lanes 16-31 -> K = k0+2,k0+3
      const int kk   = k0 + (lhi << 1);
      const int rowA = mt * 16 + ll;
      v2f aZ, aH, bZ, bH;
      aZ.x = s_aggz[rowA * AGG_PITCH + kk];
      aZ.y = s_aggz[rowA * AGG_PITCH + kk + 1];
      aH.x = s_aggh[rowA * AGG_PITCH + kk];
      aH.y = s_aggh[rowA * AGG_PITCH + kk + 1];
      // B 4x16 f32: rows K striped across lanes, column = col
      bZ.x = lin_z_w[(kk + 0) * EMB + col];
      bZ.y = lin_z_w[(kk + 1) * EMB + col];
      bH.x = lin_h_w[(kk + 0) * EMB + col];
      bH.y = lin_h_w[(kk + 1) * EMB + col];
      accZ = __builtin_amdgcn_wmma_f32_16x16x4_f32(false, aZ, false, bZ, (short)0, accZ, false, false);
      accH = __builtin_amdgcn_wmma_f32_16x16x4_f32(false, aH, false, bH, (short)0, accH, false, false);
    }
    // epilogue: gates + attention-weighted accumulation into hacc[b,n,e]
    const float bz = lin_z_b[col];
    const float bh = lin_h_b[col];
#pragma unroll
    for (int r = 0; r < 8; ++r) {
      const int row = mt * 16 + r + (lhi << 3);   // C/D: VGPR r -> M=r (lo) / M=r+8 (hi)
      if (row < NN) {
        float Z  = 1.0f / (1.0f + __expf(-(accZ[r] + bz)));
        float Ht = tanhf(accH[r] + bh);
        atomicAdd(&hacc[(b * NN + row) * EMB + col], pw * (1.0f - Z) * Ht);
      }
    }
  }
}

// ---------------------------------------------------------------------------
// Kernel 3: MLP head per batch element
// ---------------------------------------------------------------------------
__global__ void head_kernel(const float* __restrict__ hacc,
                            const float* __restrict__ lin1_w, const float* __restrict__ lin1_b,
                            const float* __restrict__ lin2_w, const float* __restrict__ lin2_b,
                            const float* __restrict__ lin3_w, const float* __restrict__ lin3_b,
                            float* __restrict__ out) {
  const int b = blockIdx.x;
  const int t = threadIdx.x;
  __shared__ float vs[NN];
  if (t < NN) {
    float hr[EMB];
#pragma unroll
    for (int e = 0; e < EMB; ++e)
      hr[e] = fmaxf(hacc[(b * NN + t) * EMB + e], 0.0f);  // relu
    float v = lin2_b[0];
    for (int j = 0; j < 32; ++j) {
      float a = lin1_b[j];
#pragma unroll
      for (int i = 0; i < EMB; ++i) a += hr[i] * lin1_w[i * 32 + j];
      v += fmaxf(a, 0.0f) * lin2_w[j];                    // relu + lin2
    }
    vs[t] = v * lin3_w[t];                                // lin3 partial
  }
  __syncthreads();
  if (t == 0) {
    float s = lin3_b[0];
    for (int n = 0; n < NN; ++n) s += vs[n];
    out[b] = 1.0f / (1.0f + __expf(-s));                  // sigmoid
  }
}

// ---------------------------------------------------------------------------
extern "C" void kernel_launch(void* const* d_in, const int* in_sizes, int n_in,
                              void* d_out, int out_size, void* d_ws, size_t ws_size,
                              hipStream_t stream) {
  (void)in_sizes; (void)n_in; (void)out_size; (void)ws_size;
  const float* x         = (const float*)d_in[0];
  const int*   ei        = (const int*)d_in[1];
  const float* conv_z_w  = (const float*)d_in[2];
  const float* conv_z_b  = (const float*)d_in[3];
  // d_in[4], d_in[5]: conv_r_* — R gate multiplies H==0, dead code
  const float* conv_h_w  = (const float*)d_in[6];
  const float* conv_h_b  = (const float*)d_in[7];
  const float* lin_z_w   = (const float*)d_in[8];
  const float* lin_z_b   = (const float*)d_in[9];
  // d_in[10], d_in[11]: lin_r_* — dead code
  const float* lin_h_w   = (const float*)d_in[12];
  const float* lin_h_b   = (const float*)d_in[13];
  const float* attention = (const float*)d_in[14];
  const float* lin1_w    = (const float*)d_in[15];
  const float* lin1_b    = (const float*)d_in[16];
  const float* lin2_w    = (const float*)d_in[17];
  const float* lin2_b    = (const float*)d_in[18];
  const float* lin3_w    = (const float*)d_in[19];
  const float* lin3_b    = (const float*)d_in[20];
  float* out = (float*)d_out;

  // workspace layout (floats)
  float* ws    = (float*)d_ws;
  float* hacc  = ws;                               // B*N*EMB = 52224
  float* normA = ws + BB * NN * EMB;               // 459
  int*   srcA  = (int*)(normA + ETOT);             // 459
  int*   dstA  = srcA + ETOT;                      // 459
  float* probs = (float*)(dstA + ETOT);            // 137

  prep_kernel<<<1, 512, 0, stream>>>(ei, attention, hacc, normA, srcA, dstA, probs);
  gate_kernel<<<dim3(PP, BB), 128, 0, stream>>>(x, conv_z_w, conv_z_b, conv_h_w, conv_h_b,
                                                lin_z_w, lin_z_b, lin_h_w, lin_h_b,
                                                normA, srcA, dstA, probs, hacc);
  head_kernel<<<BB, 64, 0, stream>>>(hacc, lin1_w, lin1_b, lin2_w, lin2_b,
                                     lin3_w, lin3_b, out);
}